// MolecularGraphNeuralNetwork_29446295781939
// MI455X (gfx1250) — compile-verified
//
#include <hip/hip_runtime.h>
#include <hip/hip_bf16.h>

// MI455X / gfx1250, wave32. FP32 WMMA (V_WMMA_F32_16X16X4_F32) for all GEMMs.

typedef float v2f __attribute__((ext_vector_type(2)));
typedef float v8f __attribute__((ext_vector_type(8)));

// Generic tiled WMMA GEMM: D[m][n] = relu?( scale * sum_k A[m][k]*B[k][n] + bias[n] + Cadd[m][n] )
// A is row-major with leading dim lda (k-stride 1).
// B element (k,n) at B[k*sbk + n*sbn]  (covers transposed weight reads).
// Mm, Nn multiples of 16; Kk multiple of 4. Whole block (8 wave32s) cooperates.
// NOTE: callers using Cadd or bias pass scale == 1.0f.
static __device__ __forceinline__ void mm_wmma(
    const float* __restrict__ A, int lda,
    const float* __restrict__ B, int sbk, int sbn,
    float* __restrict__ D, int ldd,
    int Mm, int Nn, int Kk,
    const float* __restrict__ bias,
    const float* __restrict__ Cadd, int ldc,
    float scale, bool relu)
{
    const int lane   = threadIdx.x & 31;
    const int wave   = threadIdx.x >> 5;
    const int nwaves = blockDim.x >> 5;
    const int tilesN = Nn >> 4;
    const int ntiles = (Mm >> 4) * tilesN;
    const int l15    = lane & 15;
    const int khalf  = (lane >> 4) << 1;   // 0 for lanes 0-15, 2 for lanes 16-31

    for (int t = wave; t < ntiles; t += nwaves) {
        const int tM = (t / tilesN) << 4;
        const int tN = (t % tilesN) << 4;
        const int row0 = tM + ((lane >> 4) << 3);  // C/D: VGPR r -> row0 + r
        const int col  = tN + l15;

        v8f acc = {0.f, 0.f, 0.f, 0.f, 0.f, 0.f, 0.f, 0.f};
        if (Cadd) {
#pragma unroll
            for (int r = 0; r < 8; ++r) acc[r] = Cadd[(row0 + r) * ldc + col];
        }

        const float* Ar = A + (tM + l15) * lda;
        for (int k0 = 0; k0 < Kk; k0 += 4) {
            const int kk = k0 + khalf;
            v2f a, b;
            a[0] = Ar[kk];
            a[1] = Ar[kk + 1];
            b[0] = B[kk * sbk + col * sbn];
            b[1] = B[(kk + 1) * sbk + col * sbn];
            acc = __builtin_amdgcn_wmma_f32_16x16x4_f32(
                false, a, false, b, (short)0, acc, false, false);
        }

#pragma unroll
        for (int r = 0; r < 8; ++r) {
            float val = acc[r] * scale;
            if (bias) val += bias[col];
            if (relu) val = fmaxf(val, 0.f);
            D[(row0 + r) * ldd + col] = val;
        }
    }
}

// Row-wise layernorm over 64 features: dst = (x-m)/sqrt(var+1e-5)*s + b
static __device__ __forceinline__ void ln_rows(
    const float* __restrict__ src, int lds, float* __restrict__ dst, int ldd,
    const float* __restrict__ s, const float* __restrict__ b, int nrows)
{
    int t = threadIdx.x;
    if (t < nrows) {
        const float* r = src + t * lds;
        float m = 0.f;
        for (int c = 0; c < 64; ++c) m += r[c];
        m *= (1.f / 64.f);
        float v = 0.f;
        for (int c = 0; c < 64; ++c) { float d = r[c] - m; v += d * d; }
        v *= (1.f / 64.f);
        float inv = rsqrtf(v + 1e-5f);
        float* o = dst + t * ldd;
        for (int c = 0; c < 64; ++c) o[c] = (r[c] - m) * inv * s[c] + b[c];
    }
}

// -------- Kernel 1: per-molecule GNN (embed gather + 3 layers + pool + fc) ------
// One workgroup per molecule. Exploits block-diagonal adjacency: only the
// 64x64 diagonal block is read (2 MB total instead of 268 MB dense).
__global__ __launch_bounds__(256) void gnn_fused_kernel(
    const int* __restrict__ fingerprints,    // [8192]
    const float* __restrict__ adjacency,     // [8192,8192] block-diagonal
    const float* __restrict__ embed,         // [10000,64]
    const float* __restrict__ Wfp,           // [3,64,64]
    const float* __restrict__ bfp,           // [3,64]
    const float* __restrict__ fcw,           // [64,64]
    const float* __restrict__ fcb,           // [64]
    float* __restrict__ vout)                // [128,64]
{
    __shared__ float xs[64 * 65];
    __shared__ float hs[64 * 65];
    __shared__ float as_[64 * 65];
    __shared__ float mol[64];

    const int b   = blockIdx.x;
    const int tid = threadIdx.x;

    // embed gather + diagonal adjacency block load
    for (int idx = tid; idx < 64 * 64; idx += 256) {
        const int i = idx >> 6, d = idx & 63;
        xs[i * 65 + d]  = embed[(size_t)fingerprints[b * 64 + i] * 64 + d];
        as_[i * 65 + d] = adjacency[(size_t)(b * 64 + i) * 8192 + (size_t)b * 64 + d];
    }
    __syncthreads();

    for (int l = 0; l < 3; ++l) {
        // h = relu(x @ Wfp[l]^T + bfp[l]);  B(k,n) = Wfp[l][n*64+k]
        mm_wmma(xs, 65, Wfp + l * 4096, 1, 64, hs, 65,
                64, 64, 64, bfp + l * 64, nullptr, 0, 1.f, true);
        __syncthreads();
        // x = h + A_b @ h   (C-init from h)
        mm_wmma(as_, 65, hs, 65, 1, xs, 65,
                64, 64, 64, nullptr, hs, 65, 1.f, false);
        __syncthreads();
        // row normalize: x = x / max(||x||, 1e-12)
        if (tid < 64) {
            float ss = 0.f;
            for (int d = 0; d < 64; ++d) { float t = xs[tid * 65 + d]; ss += t * t; }
            float inv = 1.f / fmaxf(sqrtf(ss), 1e-12f);
            for (int d = 0; d < 64; ++d) xs[tid * 65 + d] *= inv;
        }
        __syncthreads();
    }

    // pool (sum over 64 atoms)
    if (tid < 64) {
        float s = 0.f;
        for (int i = 0; i < 64; ++i) s += xs[i * 65 + tid];
        mol[tid] = s;
    }
    __syncthreads();
    // fc: v = mol @ fc_w^T + fc_b
    if (tid < 64) {
        float s = fcb[tid];
        for (int d = 0; d < 64; ++d) s += mol[d] * fcw[tid * 64 + d];
        vout[b * 64 + tid] = s;
    }
}

// -------- Kernel 2: transformer (2 layers) + output MLP, single workgroup ------
__global__ __launch_bounds__(256) void transformer_head_kernel(
    const float* __restrict__ vws,    // [128,64] from kernel 1
    float* __restrict__ qkv,          // ws [128,192]
    float* __restrict__ scr,          // ws [128,128]
    float* __restrict__ obuf,         // ws [128,64]
    float* __restrict__ rbuf,         // ws [128,64]
    const float* __restrict__ qkvw, const float* __restrict__ qkvb,
    const float* __restrict__ outw, const float* __restrict__ outb,
    const float* __restrict__ ln1s, const float* __restrict__ ln1b,
    const float* __restrict__ ff1w, const float* __restrict__ ff1b,
    const float* __restrict__ ff2w, const float* __restrict__ ff2b,
    const float* __restrict__ ln2s, const float* __restrict__ ln2b,
    const float* __restrict__ Woutw, const float* __restrict__ Woutb,
    const float* __restrict__ Wpw,  const float* __restrict__ Wpb,
    float* __restrict__ out)          // [128,2]
{
    __shared__ float vbuf[128 * 65];
    const int tid = threadIdx.x;

    for (int idx = tid; idx < 128 * 64; idx += 256)
        vbuf[(idx >> 6) * 65 + (idx & 63)] = vws[idx];
    __syncthreads();

    for (int l = 0; l < 2; ++l) {
        // qkv = v @ qkv_w[l]^T + qkv_b[l] : [128,192]
        mm_wmma(vbuf, 65, qkvw + l * 192 * 64, 1, 64, qkv, 192,
                128, 192, 64, qkvb + l * 192, nullptr, 0, 1.f, false);
        __syncthreads();

        for (int h = 0; h < 4; ++h) {
            // scores = (Q_h @ K_h^T) / 4 : [128,128], K=16
            mm_wmma(qkv + h * 16, 192, qkv + 64 + h * 16, 1, 192, scr, 128,
                    128, 128, 16, nullptr, nullptr, 0, 0.25f, false);
            __syncthreads();
            if (tid < 128) {  // row softmax
                float* row = scr + tid * 128;
                float mx = row[0];
                for (int j = 1; j < 128; ++j) mx = fmaxf(mx, row[j]);
                float s = 0.f;
                for (int j = 0; j < 128; ++j) { float e = __expf(row[j] - mx); row[j] = e; s += e; }
                float inv = 1.f / s;
                for (int j = 0; j < 128; ++j) row[j] *= inv;
            }
            __syncthreads();
            // O_h = attn @ V_h : [128,16], K=128
            mm_wmma(scr, 128, qkv + 128 + h * 16, 192, 1, obuf + h * 16, 64,
                    128, 16, 128, nullptr, nullptr, 0, 1.f, false);
            __syncthreads();
        }

        // rbuf = v + O @ out_w^T + out_b   (residual via C-init)
        mm_wmma(obuf, 64, outw + l * 4096, 1, 64, rbuf, 64,
                128, 64, 64, outb + l * 64, vbuf, 65, 1.f, false);
        __syncthreads();
        // LN1 -> vbuf
        ln_rows(rbuf, 64, vbuf, 65, ln1s + l * 64, ln1b + l * 64, 128);
        __syncthreads();

        // FF with residual: rbuf = v + ff2_b, then accumulate chunks of 128 over DFF=2048
        for (int idx = tid; idx < 128 * 64; idx += 256)
            rbuf[idx] = vbuf[(idx >> 6) * 65 + (idx & 63)] + ff2b[l * 64 + (idx & 63)];
        __syncthreads();
        for (int c = 0; c < 2048; c += 128) {
            // scr = relu(v @ ff1_chunk^T + b1_chunk) : [128,128]
            mm_wmma(vbuf, 65, ff1w + l * 2048 * 64 + c * 64, 1, 64, scr, 128,
                    128, 128, 64, ff1b + l * 2048 + c, nullptr, 0, 1.f, true);
            __syncthreads();
            // rbuf += scr @ ff2_chunk^T : [128,64], K=128
            mm_wmma(scr, 128, ff2w + l * 64 * 2048 + c, 1, 2048, rbuf, 64,
                    128, 64, 128, nullptr, rbuf, 64, 1.f, false);
            __syncthreads();
        }
        // LN2 -> vbuf
        ln_rows(rbuf, 64, vbuf, 65, ln2s + l * 64, ln2b + l * 64, 128);
        __syncthreads();
    }

    // output MLP: 2 relu layers
    mm_wmma(vbuf, 65, Woutw, 1, 64, obuf, 64,
            128, 64, 64, Woutb, nullptr, 0, 1.f, true);
    __syncthreads();
    mm_wmma(obuf, 64, Woutw + 4096, 1, 64, rbuf, 64,
            128, 64, 64, Woutb + 64, nullptr, 0, 1.f, true);
    __syncthreads();

    // logits = v @ Wprop^T + b : [128,2]
    {
        const int i = tid >> 1, c = tid & 1;
        float s = Wpb[c];
        for (int d = 0; d < 64; ++d) s += rbuf[i * 64 + d] * Wpw[c * 64 + d];
        out[i * 2 + c] = s;
    }
}

extern "C" void kernel_launch(void* const* d_in, const int* in_sizes, int n_in,
                              void* d_out, int out_size, void* d_ws, size_t ws_size,
                              hipStream_t stream) {
    (void)in_sizes; (void)n_in; (void)out_size; (void)ws_size;
    const int*   fingerprints = (const int*)d_in[0];
    const float* adjacency    = (const float*)d_in[1];
    const float* embed        = (const float*)d_in[2];
    const float* Wfp          = (const float*)d_in[3];
    const float* bfp          = (const float*)d_in[4];
    const float* fc_w         = (const float*)d_in[5];
    const float* fc_b         = (const float*)d_in[6];
    const float* qkv_w        = (const float*)d_in[7];
    const float* qkv_b        = (const float*)d_in[8];
    const float* out_w        = (const float*)d_in[9];
    const float* out_b        = (const float*)d_in[10];
    const float* ln1_s        = (const float*)d_in[11];
    const float* ln1_b        = (const float*)d_in[12];
    const float* ff1_w        = (const float*)d_in[13];
    const float* ff1_b        = (const float*)d_in[14];
    const float* ff2_w        = (const float*)d_in[15];
    const float* ff2_b        = (const float*)d_in[16];
    const float* ln2_s        = (const float*)d_in[17];
    const float* ln2_b        = (const float*)d_in[18];
    const float* Wout_w       = (const float*)d_in[19];
    const float* Wout_b       = (const float*)d_in[20];
    const float* Wprop_w      = (const float*)d_in[21];
    const float* Wprop_b      = (const float*)d_in[22];

    float* ws    = (float*)d_ws;
    float* vws   = ws;                    //  8192 floats
    float* qkvws = vws + 8192;            // 24576 floats
    float* scrws = qkvws + 24576;         // 16384 floats
    float* obws  = scrws + 16384;         //  8192 floats
    float* rbws  = obws + 8192;           //  8192 floats  (total 256 KB)

    gnn_fused_kernel<<<128, 256, 0, stream>>>(
        fingerprints, adjacency, embed, Wfp, bfp, fc_w, fc_b, vws);

    transformer_head_kernel<<<1, 256, 0, stream>>>(
        vws, qkvws, scrws, obws, rbws,
        qkv_w, qkv_b, out_w, out_b, ln1_s, ln1_b,
        ff1_w, ff1_b, ff2_w, ff2_b, ln2_s, ln2_b,
        Wout_w, Wout_b, Wprop_w, Wprop_b, (float*)d_out);
}